// TripletSemihardLoss_10196252361363
// MI455X (gfx1250) — compile-verified
//
#include <hip/hip_runtime.h>
#include <hip/hip_bf16.h>

typedef __attribute__((ext_vector_type(2))) float v2f;
typedef __attribute__((ext_vector_type(8))) float v8f;

#define NN 384   // number of embeddings
#define CC 256   // embedding dim
#define MARGIN_F 1.0f

// ---------------------------------------------------------------------------
// Kernel 1: G = E * E^T via V_WMMA_F32_16X16X4_F32.
// One wave (32 threads) per 16x16 output tile; grid = (24, 24).
//
// f32 A-matrix 16x4 layout (ISA 7.12.2): lane L holds row m = L%16;
//   VGPR0 = A[m][kbase], VGPR1 = A[m][kbase+1], kbase = (L/16)*2.
// B (4x16 = E^T tile) mirrors this with n = L%16, so for a Gram matrix both
// operands use the identical gather from E with different row bases.
// C/D layout: lane L holds column n = L%16, VGPR v holds row m = v + 8*(L/16).
// ---------------------------------------------------------------------------
__global__ __launch_bounds__(32)
void gram_wmma_kernel(const float* __restrict__ E, float* __restrict__ G) {
    const int ti   = blockIdx.x;          // row tile
    const int tj   = blockIdx.y;          // col tile
    const int lane = threadIdx.x;         // 0..31
    const int half = lane >> 4;           // 0 or 1
    const int r    = lane & 15;           // m for A, n for B

    const float* __restrict__ arow = E + (size_t)(ti * 16 + r) * CC;
    const float* __restrict__ brow = E + (size_t)(tj * 16 + r) * CC;
    const int koff = half * 2;

    v8f c0 = {};
    v8f c1 = {};
    for (int k0 = 0; k0 < CC; k0 += 8) {
        int ka = k0 + koff;
        v2f a0 = { arow[ka],     arow[ka + 1] };
        v2f b0 = { brow[ka],     brow[ka + 1] };
        c0 = __builtin_amdgcn_wmma_f32_16x16x4_f32(
                 false, a0, false, b0, (short)0, c0, false, false);
        int kb = k0 + 4 + koff;
        v2f a1 = { arow[kb],     arow[kb + 1] };
        v2f b1 = { brow[kb],     brow[kb + 1] };
        c1 = __builtin_amdgcn_wmma_f32_16x16x4_f32(
                 false, a1, false, b1, (short)0, c1, false, false);
    }

    float* __restrict__ gcol = G + (size_t)(tj * 16 + r);  // column n
    #pragma unroll
    for (int v = 0; v < 8; ++v) {
        int m = ti * 16 + v + 8 * half;
        gcol[(size_t)m * NN] = c0[v] + c1[v];
    }
}

// ---------------------------------------------------------------------------
// Kernel 2: one block per row i (384 threads, thread t == column j).
// Builds pd row in LDS, computes rowmin and neg_inside (reference's masked-
// maximum formula), then each positive-pair thread scans k for the semi-hard
// negative. Emits deterministic per-row loss/count partials.
// ---------------------------------------------------------------------------
__global__ __launch_bounds__(NN)
void row_loss_kernel(const float* __restrict__ G, const int* __restrict__ lab,
                     float* __restrict__ rowLoss, float* __restrict__ rowCnt) {
    __shared__ float s_pd[NN];
    __shared__ int   s_lab[NN];
    __shared__ float s_red[NN];
    __shared__ float s_rowmin;
    __shared__ float s_negin;

    const int i  = blockIdx.x;
    const int t  = threadIdx.x;
    const int li = lab[i];

    s_lab[t] = lab[t];

    // pdist(i, t) from Gram matrix
    const float gii = G[(size_t)i * NN + i];
    const float gtt = G[(size_t)t * NN + t];
    const float git = G[(size_t)i * NN + t];
    float pd2 = gii + gtt - 2.0f * git;
    float pd  = (pd2 > 0.0f) ? pd2 : 0.0f;   // clip + error-mask combined
    if (t == i) pd = 0.0f;                   // zero diagonal
    s_pd[t] = pd;
    __syncthreads();

    // ---- rowmin = min_k pd[i][k] ----
    s_red[t] = pd;
    __syncthreads();
    for (int s = 256; s > 0; s >>= 1) {
        if (t < s && t + s < NN) s_red[t] = fminf(s_red[t], s_red[t + s]);
        __syncthreads();
    }
    if (t == 0) s_rowmin = s_red[0];
    __syncthreads();
    const float rowmin = s_rowmin;

    // ---- neg_inside = max_k ((pd - rowmin) * adj_not) + rowmin ----
    const bool adjnot_t = (s_lab[t] != li);
    s_red[t] = adjnot_t ? (pd - rowmin) : 0.0f;
    __syncthreads();
    for (int s = 256; s > 0; s >>= 1) {
        if (t < s && t + s < NN) s_red[t] = fmaxf(s_red[t], s_red[t + s]);
        __syncthreads();
    }
    if (t == 0) s_negin = s_red[0] + rowmin;
    __syncthreads();
    const float neg_inside = s_negin;

    // ---- per positive pair: semi-hard negative scan ----
    float loss = 0.0f;
    float cnt  = 0.0f;
    const bool positive = (s_lab[t] == li) && (t != i);
    if (positive) {
        const float pdij = pd;
        float minOut = 3.402823466e38f;
        bool  exists = false;
        for (int k = 0; k < NN; ++k) {
            const float pdk = s_pd[k];
            if ((s_lab[k] != li) && (pdk > pdij)) {
                exists = true;
                minOut = fminf(minOut, pdk);
            }
        }
        const float shn = exists ? minOut : neg_inside;
        const float l   = MARGIN_F + pdij - shn;
        loss = (l > 0.0f) ? l : 0.0f;
        cnt  = 1.0f;
    }

    // ---- per-row sums (deterministic; no atomics) ----
    s_red[t] = loss;
    __syncthreads();
    for (int s = 256; s > 0; s >>= 1) {
        if (t < s && t + s < NN) s_red[t] += s_red[t + s];
        __syncthreads();
    }
    if (t == 0) rowLoss[i] = s_red[0];
    __syncthreads();

    s_red[t] = cnt;
    __syncthreads();
    for (int s = 256; s > 0; s >>= 1) {
        if (t < s && t + s < NN) s_red[t] += s_red[t + s];
        __syncthreads();
    }
    if (t == 0) rowCnt[i] = s_red[0];
}

// ---------------------------------------------------------------------------
// Kernel 3: final reduction + division. One block, 384 threads.
// ---------------------------------------------------------------------------
__global__ __launch_bounds__(NN)
void finalize_kernel(const float* __restrict__ rowLoss,
                     const float* __restrict__ rowCnt,
                     float* __restrict__ out) {
    __shared__ float sl[NN];
    __shared__ float sc[NN];
    const int t = threadIdx.x;
    sl[t] = rowLoss[t];
    sc[t] = rowCnt[t];
    __syncthreads();
    for (int s = 256; s > 0; s >>= 1) {
        if (t < s && t + s < NN) { sl[t] += sl[t + s]; sc[t] += sc[t + s]; }
        __syncthreads();
    }
    if (t == 0) out[0] = sl[0] / sc[0];
}

// ---------------------------------------------------------------------------
extern "C" void kernel_launch(void* const* d_in, const int* in_sizes, int n_in,
                              void* d_out, int out_size, void* d_ws, size_t ws_size,
                              hipStream_t stream) {
    const float* E   = (const float*)d_in[0];   // [384, 256] f32
    const int*   lab = (const int*)d_in[1];     // [384] int
    float*       out = (float*)d_out;           // scalar f32

    float* G       = (float*)d_ws;              // 384*384 floats
    float* rowLoss = G + (size_t)NN * NN;       // 384 floats
    float* rowCnt  = rowLoss + NN;              // 384 floats

    dim3 gridG(NN / 16, NN / 16);
    gram_wmma_kernel<<<gridG, 32, 0, stream>>>(E, G);
    row_loss_kernel<<<NN, NN, 0, stream>>>(G, lab, rowLoss, rowCnt);
    finalize_kernel<<<1, NN, 0, stream>>>(rowLoss, rowCnt, out);
}